// MicroTopoBrain_18769007084089
// MI455X (gfx1250) — compile-verified
//
#include <hip/hip_runtime.h>
#include <hip/hip_bf16.h>

// MicroTopoBrain on MI455X (gfx1250), fp32 end-to-end.
// Structure:
//   setup_kernel : fold adj-mix into embed (W', b'), precompute K = basis@k_w+k_b, ||Ww||_F,
//                  and the constant part of the homeostatic MLP first layer (h1_base)
//   pass1_kernel : WMMA f32 16x16x4 embed -> per-node-row homeostatic MLP -> per-block partials
//   reduce_kernel: deterministic tree reduction -> gate_value
//   pass2_kernel : WMMA embed -> cell/gate/basis-attention per row -> WMMA readout (32->3, N padded)
//
// Round-2 fix: waves-per-EU=4 (256-VGPR cap) caused scratch spill/reload inside the hot
// loop (compiler hoists ~350 LDS-resident weight floats into VGPRs). Instead:
//   - pass1: cap 512 VGPRs (2 waves/SIMD) -> no spills; VALU-dense loop hides latency anyway.
//   - pass2: smaller weight set -> cap ~336 (3 waves/SIMD).
//   - fold constant MLP inputs (wnorm, 0.5) into h1_base and stage only aw[:,2]/ab[2],
//     trimming both hoisted registers (-72 floats) and ~134M FMAs.

typedef __attribute__((ext_vector_type(2))) float v2f;
typedef __attribute__((ext_vector_type(8))) float v8f;

#define B_ROWS     1048576
#define TILES      (B_ROWS / 16)
#define BLOCK      256
#define WPB        (BLOCK / 32)      // waves per block (wave32)
#define GRID1      1024
#define GRID2      2048
#define LDS_STRIDE 34                // 16x32 tile, stride 34 avoids bank conflicts on D-layout stores
#define TILE_F     (16 * LDS_STRIDE)

// workspace layout (floats)
#define WS_WP      0                 // 12x32 folded embed weight  W' = W_embed @ M_blk
#define WS_BP      384               // 32 folded embed bias
#define WS_K       416               // 2x8  K = basis @ k_w + k_b
#define WS_WNORM   432               // ||Ww||_F
#define WS_GATE    433               // gate_value (written by reduce_kernel)
#define WS_H1B     440               // 16  h1_base = b1 + wnorm*w1[2,:] + 0.5*w1[3,:]
#define WS_PART    464               // GRID1 per-block partials

__device__ __forceinline__ float sigmoidf_fast(float x) {
  x = fminf(fmaxf(x, -30.f), 30.f);
  return 1.f / (1.f + __expf(-x));
}
__device__ __forceinline__ float tanhf_fast(float x) {
  x = fminf(fmaxf(x, -15.f), 15.f);
  float e = __expf(2.f * x);
  return (e - 1.f) / (e + 1.f);
}
__device__ __forceinline__ float clip2(float v) { return fminf(fmaxf(v, -2.f), 2.f); }
__device__ __forceinline__ float clip5(float v) { return fminf(fmaxf(v, -5.f), 5.f); }

// D = A(16x4 f32) x B(4x16 f32) + C(16x16 f32), full fp32 precision.
__device__ __forceinline__ v8f wmma_f32_k4(v2f a, v2f b, v8f c) {
  return __builtin_amdgcn_wmma_f32_16x16x4_f32(false, a, false, b, (short)0, c, false, false);
}

// ---------------------------------------------------------------------------
__global__ void setup_kernel(const float* __restrict__ W_embed, const float* __restrict__ b_embed,
                             const float* __restrict__ adj_w, const float* __restrict__ Ww,
                             const float* __restrict__ basis, const float* __restrict__ k_w,
                             const float* __restrict__ k_b, const float* __restrict__ w1,
                             const float* __restrict__ b1, float* __restrict__ ws) {
  if (blockIdx.x != 0 || threadIdx.x != 0) return;
  // 2x2 grid neighbors (MASK): node i mixes from exactly 2 neighbors.
  const int nb[4][2] = { {1, 2}, {0, 3}, {0, 3}, {1, 2} };
  float a[4][2];
  for (int i = 0; i < 4; ++i) {
    float a0 = 1.f / (1.f + __expf(-adj_w[i * 4 + nb[i][0]]));
    float a1 = 1.f / (1.f + __expf(-adj_w[i * 4 + nb[i][1]]));
    float rs = fmaxf(a0 + a1, 1e-6f);
    a[i][0] = a0 / rs;
    a[i][1] = a1 / rs;
  }
  // W'[k, i*8+d] = sum_j adj[i,j] * W_embed[k, j*8+d]  (fold topology mix into embed)
  for (int k = 0; k < 12; ++k)
    for (int i = 0; i < 4; ++i)
      for (int d = 0; d < 8; ++d)
        ws[WS_WP + k * 32 + i * 8 + d] =
            a[i][0] * W_embed[k * 32 + nb[i][0] * 8 + d] +
            a[i][1] * W_embed[k * 32 + nb[i][1] * 8 + d];
  for (int i = 0; i < 4; ++i)
    for (int d = 0; d < 8; ++d)
      ws[WS_BP + i * 8 + d] = a[i][0] * b_embed[nb[i][0] * 8 + d] +
                              a[i][1] * b_embed[nb[i][1] * 8 + d];
  // K = basis @ k_w + k_b  (2x8)
  for (int t = 0; t < 2; ++t)
    for (int d = 0; d < 8; ++d) {
      float acc = k_b[d];
      for (int k = 0; k < 8; ++k) acc += basis[t * 8 + k] * k_w[k * 8 + d];
      ws[WS_K + t * 8 + d] = acc;
    }
  float acc = 0.f;
  for (int i = 0; i < 64; ++i) acc += Ww[i] * Ww[i];
  float wnorm = sqrtf(acc);
  ws[WS_WNORM] = wnorm;
  // constant part of homeostatic layer 1: inputs s2=wnorm, s3=0.5 are row-invariant
  for (int j = 0; j < 16; ++j)
    ws[WS_H1B + j] = b1[j] + wnorm * w1[32 + j] + 0.5f * w1[48 + j];
}

// ---------------------------------------------------------------------------
// Shared embed: one wave computes a 16x32 h-tile via 6 f32 WMMAs and parks it in LDS.
// Fragment layouts (ISA 7.12.2): row = vgpr_idx + (rows/2)*lane_half.
//   A 16x4 : lane l -> m=l&15, half h=l>>4 : a.{x,y} = A[m][k0+2h+{0,1}]
//   B 4x16 : lane l -> n=l&15, half h=l>>4 : b.{x,y} = B[k0+2h+{0,1}][n]
//   D 16x16: vgpr r, lane l -> M=r+8h, N=l&15
// ---------------------------------------------------------------------------

__global__ __launch_bounds__(BLOCK, 2) void pass1_kernel(
    const float* __restrict__ x, const float* __restrict__ Ww,
    const float* __restrict__ w1, const float* __restrict__ g,
    const float* __restrict__ be, const float* __restrict__ w2,
    const float* __restrict__ b2, const float* __restrict__ aw,
    const float* __restrict__ ab, const float* __restrict__ ws,
    float* __restrict__ part) {
  __shared__ float wb[292];
  __shared__ float tileS[WPB][TILE_F];
  __shared__ float red[BLOCK];
  // LDS-staged weight offsets (trimmed hot set: 289 floats)
  const int WW_O = 0, W1_O = 64, H1B_O = 96, G_O = 112, BE_O = 128,
            W2_O = 144, B2_O = 272, AW2_O = 280, AB2_O = 288;

  const int tid = threadIdx.x;
  for (int i = tid; i < 64; i += BLOCK) wb[WW_O + i] = Ww[i];
  for (int i = tid; i < 32; i += BLOCK) wb[W1_O + i] = w1[i];            // rows 0,1 only
  for (int i = tid; i < 16; i += BLOCK) {
    wb[H1B_O + i] = ws[WS_H1B + i];
    wb[G_O + i] = g[i];
    wb[BE_O + i] = be[i];
  }
  for (int i = tid; i < 128; i += BLOCK) wb[W2_O + i] = w2[i];
  for (int i = tid; i < 8; i += BLOCK) { wb[B2_O + i] = b2[i]; wb[AW2_O + i] = aw[i * 4 + 2]; }
  if (tid == 0) wb[AB2_O] = ab[2];
  __syncthreads();

  const int lane = tid & 31, wv = tid >> 5;
  const int nn = lane & 15, hh = lane >> 4;

  // Hoist loop-invariant B fragments for the embed GEMM (K=12 -> 3 steps, N=32 -> 2 tiles)
  const float* Wp = ws + WS_WP;
  v2f bfr[3][2];
#pragma unroll
  for (int ks = 0; ks < 3; ++ks)
#pragma unroll
    for (int nt = 0; nt < 2; ++nt) {
      int k = ks * 4 + 2 * hh;
      bfr[ks][nt] = (v2f){ Wp[k * 32 + nt * 16 + nn], Wp[(k + 1) * 32 + nt * 16 + nn] };
    }
  const float bp0 = ws[WS_BP + nn], bp1 = ws[WS_BP + 16 + nn];

  float* T = &tileS[wv][0];
  const int waveId = blockIdx.x * WPB + wv;
  const int nwaves = GRID1 * WPB;
  float local = 0.f;

  for (int t = waveId; t < TILES; t += nwaves) {
    const float* xr = x + (long)(t * 16 + nn) * 12;
    if (t + nwaves < TILES)
      __builtin_prefetch(x + (long)((t + nwaves) * 16 + nn) * 12, 0, 1);
    v2f a0 = { xr[2 * hh], xr[2 * hh + 1] };
    v2f a1 = { xr[4 + 2 * hh], xr[5 + 2 * hh] };
    v2f a2 = { xr[8 + 2 * hh], xr[9 + 2 * hh] };
    v8f acc0 = { bp0, bp0, bp0, bp0, bp0, bp0, bp0, bp0 };
    v8f acc1 = { bp1, bp1, bp1, bp1, bp1, bp1, bp1, bp1 };
    acc0 = wmma_f32_k4(a0, bfr[0][0], acc0);
    acc0 = wmma_f32_k4(a1, bfr[1][0], acc0);
    acc0 = wmma_f32_k4(a2, bfr[2][0], acc0);
    acc1 = wmma_f32_k4(a0, bfr[0][1], acc1);
    acc1 = wmma_f32_k4(a1, bfr[1][1], acc1);
    acc1 = wmma_f32_k4(a2, bfr[2][1], acc1);
#pragma unroll
    for (int r = 0; r < 8; ++r) {
      T[(r + 8 * hh) * LDS_STRIDE + nn] = acc0[r];
      T[(r + 8 * hh) * LDS_STRIDE + 16 + nn] = acc1[r];
    }
    // 64 node-rows per tile; each lane takes 2 (same-wave LDS is in-order, no barrier needed)
#pragma unroll
    for (int rr = 0; rr < 2; ++rr) {
      const float* hp = T + ((lane >> 2) + rr * 8) * LDS_STRIDE + (lane & 3) * 8;
      float h[8];
#pragma unroll
      for (int d = 0; d < 8; ++d) h[d] = hp[d];
      float sm = 0.f, sq = 0.f;
#pragma unroll
      for (int d = 0; d < 8; ++d) { sm += h[d]; sq += h[d] * h[d]; }
      float mean = sm * 0.125f;
      float var = sq * 0.125f - mean * mean;
      float s0 = fabsf(var - 0.5f);                 // surprise
      float ex = 0.f;
#pragma unroll
      for (int d = 0; d < 8; ++d) {
        float a = 0.f;
#pragma unroll
        for (int k = 0; k < 8; ++k) a += h[k] * wb[WW_O + k * 8 + d];
        ex += fabsf(a);
      }
      float s1 = ex * 0.125f;                       // excitation
      // layer 1: h1 = h1_base + s0*w1[0,:] + s1*w1[1,:]  -> LN -> tanh
      float h1[16];
      float m1 = 0.f;
#pragma unroll
      for (int j = 0; j < 16; ++j) {
        float a = wb[H1B_O + j] + s0 * wb[W1_O + j] + s1 * wb[W1_O + 16 + j];
        h1[j] = a;
        m1 += a;
      }
      m1 *= (1.f / 16.f);
      float v1 = 0.f;
#pragma unroll
      for (int j = 0; j < 16; ++j) { float d = h1[j] - m1; v1 += d * d; }
      v1 *= (1.f / 16.f);
      float inv = rsqrtf(v1 + 1e-5f);
#pragma unroll
      for (int j = 0; j < 16; ++j)
        h1[j] = tanhf_fast((h1[j] - m1) * inv * wb[G_O + j] + wb[BE_O + j]);
      // 16 -> 8 relu -> sigmoid head column 2
      float zc = wb[AB2_O];
#pragma unroll
      for (int d = 0; d < 8; ++d) {
        float a = wb[B2_O + d];
#pragma unroll
        for (int j = 0; j < 16; ++j) a += h1[j] * wb[W2_O + j * 8 + d];
        a = fmaxf(a, 0.f);
        zc += a * wb[AW2_O + d];
      }
      local += sigmoidf_fast(zc);
    }
  }
  // deterministic per-block reduction
  red[tid] = local;
  __syncthreads();
  for (int s = BLOCK / 2; s > 0; s >>= 1) {
    if (tid < s) red[tid] += red[tid + s];
    __syncthreads();
  }
  if (tid == 0) part[blockIdx.x] = red[0];
}

// ---------------------------------------------------------------------------
__global__ __launch_bounds__(GRID1) void reduce_kernel(const float* __restrict__ part,
                                                       float* __restrict__ gate) {
  __shared__ float red[GRID1];
  const int tid = threadIdx.x;
  red[tid] = part[tid];
  __syncthreads();
  for (int s = GRID1 / 2; s > 0; s >>= 1) {
    if (tid < s) red[tid] += red[tid + s];
    __syncthreads();
  }
  if (tid == 0) gate[0] = red[0] * (1.f / 4194304.f);  // mean over B*N rows
}

// ---------------------------------------------------------------------------
__global__ __launch_bounds__(BLOCK, 3) void pass2_kernel(
    const float* __restrict__ x, const float* __restrict__ Vw,
    const float* __restrict__ gate_w, const float* __restrict__ gate_b,
    const float* __restrict__ q_w, const float* __restrict__ q_b,
    const float* __restrict__ basis, const float* __restrict__ W_out,
    const float* __restrict__ b_out, const float* __restrict__ ws,
    float* __restrict__ out) {
  __shared__ float wb[184];
  __shared__ float tileS[WPB][TILE_F];
  const int VW_O = 0, QW_O = 64, QB_O = 128, BAS_O = 136, KC_O = 152, GW_O = 168, GB_O = 176;

  const int tid = threadIdx.x;
  for (int i = tid; i < 64; i += BLOCK) { wb[VW_O + i] = Vw[i]; wb[QW_O + i] = q_w[i]; }
  for (int i = tid; i < 8; i += BLOCK) { wb[QB_O + i] = q_b[i]; wb[GW_O + i] = gate_w[i]; }
  for (int i = tid; i < 16; i += BLOCK) { wb[BAS_O + i] = basis[i]; wb[KC_O + i] = ws[WS_K + i]; }
  if (tid == 0) wb[GB_O] = gate_b[0];
  __syncthreads();

  const int lane = tid & 31, wv = tid >> 5;
  const int nn = lane & 15, hh = lane >> 4;
  const float gateval = ws[WS_GATE];
  const float invT = 1.1785113019775793f;  // 1 / max(sqrt(8)*0.3, 0.5)

  // embed B fragments
  const float* Wp = ws + WS_WP;
  v2f bfr[3][2];
#pragma unroll
  for (int ks = 0; ks < 3; ++ks)
#pragma unroll
    for (int nt = 0; nt < 2; ++nt) {
      int k = ks * 4 + 2 * hh;
      bfr[ks][nt] = (v2f){ Wp[k * 32 + nt * 16 + nn], Wp[(k + 1) * 32 + nt * 16 + nn] };
    }
  const float bp0 = ws[WS_BP + nn], bp1 = ws[WS_BP + 16 + nn];

  // readout B fragments: 32x3 padded to 32x16 (K=32 -> 8 steps)
  v2f bo[8];
#pragma unroll
  for (int ks = 0; ks < 8; ++ks) {
    int k = ks * 4 + 2 * hh;
    float bx = (nn < 3) ? W_out[k * 3 + nn] : 0.f;
    float by = (nn < 3) ? W_out[(k + 1) * 3 + nn] : 0.f;
    bo[ks] = (v2f){ bx, by };
  }
  const float co = (nn < 3) ? b_out[nn] : 0.f;

  float* T = &tileS[wv][0];
  const int waveId = blockIdx.x * WPB + wv;
  const int nwaves = GRID2 * WPB;

  for (int t = waveId; t < TILES; t += nwaves) {
    const float* xr = x + (long)(t * 16 + nn) * 12;
    if (t + nwaves < TILES)
      __builtin_prefetch(x + (long)((t + nwaves) * 16 + nn) * 12, 0, 1);
    v2f a0 = { xr[2 * hh], xr[2 * hh + 1] };
    v2f a1 = { xr[4 + 2 * hh], xr[5 + 2 * hh] };
    v2f a2 = { xr[8 + 2 * hh], xr[9 + 2 * hh] };
    v8f acc0 = { bp0, bp0, bp0, bp0, bp0, bp0, bp0, bp0 };
    v8f acc1 = { bp1, bp1, bp1, bp1, bp1, bp1, bp1, bp1 };
    acc0 = wmma_f32_k4(a0, bfr[0][0], acc0);
    acc0 = wmma_f32_k4(a1, bfr[1][0], acc0);
    acc0 = wmma_f32_k4(a2, bfr[2][0], acc0);
    acc1 = wmma_f32_k4(a0, bfr[0][1], acc1);
    acc1 = wmma_f32_k4(a1, bfr[1][1], acc1);
    acc1 = wmma_f32_k4(a2, bfr[2][1], acc1);
#pragma unroll
    for (int r = 0; r < 8; ++r) {
      T[(r + 8 * hh) * LDS_STRIDE + nn] = acc0[r];
      T[(r + 8 * hh) * LDS_STRIDE + 16 + nn] = acc1[r];
    }
    // per-node-row: v -> gate -> cell -> Q -> 2-way softmax over basis -> x_out (in-place in LDS)
#pragma unroll
    for (int rr = 0; rr < 2; ++rr) {
      float* hp = T + ((lane >> 2) + rr * 8) * LDS_STRIDE + (lane & 3) * 8;
      float h[8];
#pragma unroll
      for (int d = 0; d < 8; ++d) h[d] = hp[d];
      float v[8];
#pragma unroll
      for (int d = 0; d < 8; ++d) {
        float a = 0.f;
#pragma unroll
        for (int k = 0; k < 8; ++k) a += h[k] * wb[VW_O + k * 8 + d];
        v[d] = clip2(a);
      }
      float gz = wb[GB_O];
#pragma unroll
      for (int d = 0; d < 8; ++d) gz += v[d] * wb[GW_O + d];
      float gate = sigmoidf_fast(gz) * gateval;
      float cell[8];
#pragma unroll
      for (int d = 0; d < 8; ++d) cell[d] = clip2(gate * v[d]);
      float q[8];
#pragma unroll
      for (int d = 0; d < 8; ++d) {
        float a = wb[QB_O + d];
#pragma unroll
        for (int k = 0; k < 8; ++k) a += cell[k] * wb[QW_O + k * 8 + d];
        q[d] = a;
      }
      float at0 = 0.f, at1 = 0.f;
#pragma unroll
      for (int d = 0; d < 8; ++d) { at0 += q[d] * wb[KC_O + d]; at1 += q[d] * wb[KC_O + 8 + d]; }
      at0 = clip5(at0 * invT);
      at1 = clip5(at1 * invT);
      float mx = fmaxf(at0, at1);
      float e0 = __expf(at0 - mx), e1 = __expf(at1 - mx);
      float is = 1.f / (e0 + e1);
      float w0 = e0 * is, w1s = e1 * is;
#pragma unroll
      for (int d = 0; d < 8; ++d)
        hp[d] = clip2(w0 * wb[BAS_O + d] + w1s * wb[BAS_O + 8 + d] + 0.2f * cell[d]);
    }
    // readout: x_out[16x32] @ W_out_padded[32x16] + b_out, via 8 f32 WMMAs
    v8f o = { co, co, co, co, co, co, co, co };
#pragma unroll
    for (int ks = 0; ks < 8; ++ks) {
      int k0 = ks * 4 + 2 * hh;
      v2f a = { T[nn * LDS_STRIDE + k0], T[nn * LDS_STRIDE + k0 + 1] };
      o = wmma_f32_k4(a, bo[ks], o);
    }
    if (nn < 3) {
#pragma unroll
      for (int r = 0; r < 8; ++r)
        out[(long)(t * 16 + r + 8 * hh) * 3 + nn] = o[r];
    }
  }
}

// ---------------------------------------------------------------------------
extern "C" void kernel_launch(void* const* d_in, const int* in_sizes, int n_in,
                              void* d_out, int out_size, void* d_ws, size_t ws_size,
                              hipStream_t stream) {
  (void)in_sizes; (void)n_in; (void)out_size; (void)ws_size;
  const float* x       = (const float*)d_in[0];
  const float* W_embed = (const float*)d_in[1];
  const float* b_embed = (const float*)d_in[2];
  const float* adj_w   = (const float*)d_in[3];
  const float* Vw      = (const float*)d_in[4];
  const float* Ww      = (const float*)d_in[5];
  const float* gate_w  = (const float*)d_in[6];
  const float* gate_b  = (const float*)d_in[7];
  const float* hc_w1   = (const float*)d_in[8];
  const float* hc_b1   = (const float*)d_in[9];
  const float* hc_g    = (const float*)d_in[10];
  const float* hc_be   = (const float*)d_in[11];
  const float* hc_w2   = (const float*)d_in[12];
  const float* hc_b2   = (const float*)d_in[13];
  const float* hc_aw   = (const float*)d_in[14];
  const float* hc_ab   = (const float*)d_in[15];
  const float* basis   = (const float*)d_in[16];
  const float* q_w     = (const float*)d_in[17];
  const float* q_b     = (const float*)d_in[18];
  const float* k_w     = (const float*)d_in[19];
  const float* k_b     = (const float*)d_in[20];
  const float* W_out   = (const float*)d_in[21];
  const float* b_out   = (const float*)d_in[22];
  float* out = (float*)d_out;
  float* ws  = (float*)d_ws;

  setup_kernel<<<1, 64, 0, stream>>>(W_embed, b_embed, adj_w, Ww, basis, k_w, k_b,
                                     hc_w1, hc_b1, ws);
  pass1_kernel<<<GRID1, BLOCK, 0, stream>>>(x, Ww, hc_w1, hc_g, hc_be,
                                            hc_w2, hc_b2, hc_aw, hc_ab, ws, ws + WS_PART);
  reduce_kernel<<<1, GRID1, 0, stream>>>(ws + WS_PART, ws + WS_GATE);
  pass2_kernel<<<GRID2, BLOCK, 0, stream>>>(x, Vw, gate_w, gate_b, q_w, q_b,
                                            basis, W_out, b_out, ws, out);
}